// FM_84920093376777
// MI455X (gfx1250) — compile-verified
//
#include <hip/hip_runtime.h>

typedef float v2f __attribute__((ext_vector_type(2)));
typedef float v8f __attribute__((ext_vector_type(8)));

#define FM_NUM_FEATS 3
#define FM_EMB 16
#define FM_ROWS_PER_WAVE 16
#define FM_ROWS_PER_BLOCK 128   // 8 wave32 per 256-thread block

// Accumulate one v-row (this lane's 8 of 16 emb dims, as 4 element-pairs)
// into the WMMA A-operand pairs s0..s3 and the sum-of-squares partial p.
__device__ __forceinline__ void fm_accum_row(const float* __restrict__ vr,
                                             float w, float w2,
                                             v2f& s0, v2f& s1, v2f& s2, v2f& s3,
                                             float& p) {
    v2f e0 = *(const v2f*)(vr + 0);
    v2f e1 = *(const v2f*)(vr + 4);
    v2f e2 = *(const v2f*)(vr + 8);
    v2f e3 = *(const v2f*)(vr + 12);
    s0.x += w * e0.x; s0.y += w * e0.y;
    s1.x += w * e1.x; s1.y += w * e1.y;
    s2.x += w * e2.x; s2.y += w * e2.y;
    s3.x += w * e3.x; s3.y += w * e3.y;
    float q = e0.x*e0.x + e0.y*e0.y + e1.x*e1.x + e1.y*e1.y
            + e2.x*e2.x + e2.y*e2.y + e3.x*e3.x + e3.y*e3.y;
    p += w2 * q;
}

__global__ __launch_bounds__(256, 1)
void fm_fused_kernel(const float* __restrict__ x_num,
                     const int*   __restrict__ x_cat,
                     const float* __restrict__ v,
                     const float* __restrict__ global_bias,
                     const float* __restrict__ num_bias,
                     const float* __restrict__ cat_bias,
                     float* __restrict__ out) {
    const int lane = threadIdx.x & 31;
    const int wave = threadIdx.x >> 5;
    const int r    = lane & 15;        // row slot within this wave's 16-row tile
    const int h    = lane >> 4;        // which element-pair of each K-quad this lane owns
    const int row  = blockIdx.x * FM_ROWS_PER_BLOCK + wave * FM_ROWS_PER_WAVE + r;
    const int koff = 2 * h;            // lanes 0-15: elems {4j,4j+1}; lanes 16-31: {4j+2,4j+3}

    // s[j] = this lane's pair of sum_vec elems for K-quad j (A 16x4 f32 layout)
    v2f s0 = {0.f, 0.f}, s1 = {0.f, 0.f}, s2 = {0.f, 0.f}, s3 = {0.f, 0.f};
    float p    = 0.f;                  // partial sum_of_square over this lane's 8 dims
    float bias = global_bias[0];

    // 3 numeric features: v rows 0..2, weight x_num[row][f]
    const float* xn = x_num + row * FM_NUM_FEATS;
    #pragma unroll
    for (int f = 0; f < FM_NUM_FEATS; ++f) {
        const float w = xn[f];
        bias += w * num_bias[f];
        fm_accum_row(v + f * FM_EMB + koff, w, w * w, s0, s1, s2, s3, p);
    }

    // 4 one-hot categorical features: weight 1, v row = 3 + offset + x_cat
    const int offs0 = 0, offs1 = 10000, offs2 = 18000, offs3 = 18100;
    const int* xc = x_cat + row * 4;
    {
        int g0 = xc[0] + offs0;
        int g1 = xc[1] + offs1;
        int g2 = xc[2] + offs2;
        int g3 = xc[3] + offs3;
        bias += cat_bias[g0] + cat_bias[g1] + cat_bias[g2] + cat_bias[g3];
        fm_accum_row(v + (FM_NUM_FEATS + g0) * FM_EMB + koff, 1.f, 1.f, s0, s1, s2, s3, p);
        fm_accum_row(v + (FM_NUM_FEATS + g1) * FM_EMB + koff, 1.f, 1.f, s0, s1, s2, s3, p);
        fm_accum_row(v + (FM_NUM_FEATS + g2) * FM_EMB + koff, 1.f, 1.f, s0, s1, s2, s3, p);
        fm_accum_row(v + (FM_NUM_FEATS + g3) * FM_EMB + koff, 1.f, 1.f, s0, s1, s2, s3, p);
    }

    // Combine the two half-lane partials of sum_of_square for this row (wave32 xor-16).
    p += __shfl_xor(p, 16, 32);

    // D = S * S^T via 4 chained V_WMMA_F32_16X16X4_F32 over K=16.
    // A and B register layouts for S and S^T coincide, so SRC0 == SRC1.
    // diag(D)[r] = ||sum_vec_r||^2 in exact f32.
    v8f acc = {};
    acc = __builtin_amdgcn_wmma_f32_16x16x4_f32(false, s0, false, s0, (short)0, acc, false, false);
    acc = __builtin_amdgcn_wmma_f32_16x16x4_f32(false, s1, false, s1, (short)0, acc, false, false);
    acc = __builtin_amdgcn_wmma_f32_16x16x4_f32(false, s2, false, s2, (short)0, acc, false, false);
    acc = __builtin_amdgcn_wmma_f32_16x16x4_f32(false, s3, false, s3, (short)0, acc, false, false);

    // C/D 16x16 f32 layout: row M<8 -> vgpr M @ lane M ; row M>=8 -> vgpr M-8 @ lane M+16.
    // So lanes 0-7 hold diag rows 0-7 and lanes 24-31 hold diag rows 8-15.
    const int vi = (lane < 8) ? lane : ((lane >= 24) ? (lane - 24) : 0);
    float d = acc[0];
    if (vi == 1) d = acc[1];
    if (vi == 2) d = acc[2];
    if (vi == 3) d = acc[3];
    if (vi == 4) d = acc[4];
    if (vi == 5) d = acc[5];
    if (vi == 6) d = acc[6];
    if (vi == 7) d = acc[7];

    const float y = bias + 0.5f * (d - p);
    if (lane < 8 || lane >= 24) {
        out[row] = y;   // lane<8: row = base+lane ; lane>=24: row = base+lane-16
    }
}

extern "C" void kernel_launch(void* const* d_in, const int* in_sizes, int n_in,
                              void* d_out, int out_size, void* d_ws, size_t ws_size,
                              hipStream_t stream) {
    (void)n_in; (void)out_size; (void)d_ws; (void)ws_size;
    const float* x_num = (const float*)d_in[0];
    const int*   x_cat = (const int*)d_in[1];
    const float* v     = (const float*)d_in[2];
    const float* gb    = (const float*)d_in[3];
    const float* nb    = (const float*)d_in[4];
    const float* cb    = (const float*)d_in[5];
    float* out = (float*)d_out;

    const int rows   = in_sizes[0] / FM_NUM_FEATS;          // 8192
    const int blocks = (rows + FM_ROWS_PER_BLOCK - 1) / FM_ROWS_PER_BLOCK;  // 64
    fm_fused_kernel<<<blocks, 256, 0, stream>>>(x_num, x_cat, v, gb, nb, cb, out);
}